// GNNModel_5781025980454
// MI455X (gfx1250) — compile-verified
//
#include <hip/hip_runtime.h>

// GCN (2x GCNConv + mean-pool) for MI455X gfx1250.
// Memory-bound (edge scatter ~614MB/call, fits in 192MB L2); the dense
// h1 @ W2 GEMM is done natively in fp32 with V_WMMA_F32_16X16X4_F32.

typedef __attribute__((ext_vector_type(2))) float v2f;
typedef __attribute__((ext_vector_type(8))) float v8f;

#define HID 64

// ---------------- small helper kernels ----------------

__global__ __launch_bounds__(256) void k_init(float* __restrict__ deg, float* __restrict__ s1,
                                              float* __restrict__ gsum, float* __restrict__ gcnt,
                                              int N, int G) {
  int i = blockIdx.x * blockDim.x + threadIdx.x;
  if (i < N) { deg[i] = 1.0f; s1[i] = 0.0f; }   // self-loop contributes 1 to degree
  if (i < G * HID) gsum[i] = 0.0f;
  if (i < G) gcnt[i] = 0.0f;
}

__global__ __launch_bounds__(256) void k_deg(const int* __restrict__ dst, float* __restrict__ deg, int E) {
  int e = blockIdx.x * blockDim.x + threadIdx.x;
  if (e < E) atomicAdd(&deg[dst[e]], 1.0f);
}

__global__ __launch_bounds__(256) void k_dinv(const float* __restrict__ deg, float* __restrict__ dinv, int N) {
  int i = blockIdx.x * blockDim.x + threadIdx.x;
  if (i < N) dinv[i] = rsqrtf(deg[i]);          // deg >= 1 always (self loop)
}

// Layer-1 message passing collapses to a scalar scatter because x is [N,1].
__global__ __launch_bounds__(256) void k_scatter1(const int* __restrict__ src, const int* __restrict__ dst,
                                                  const float* __restrict__ x, const float* __restrict__ dinv,
                                                  float* __restrict__ s1, int E) {
  int e = blockIdx.x * blockDim.x + threadIdx.x;
  if (e < E) {
    int s = src[e];
    atomicAdd(&s1[dst[e]], dinv[s] * x[s]);
  }
}

// h1[n][h] = relu( dinv[n]*(s1[n] + dinv[n]*x[n]) * W1[h] + b1[h] )
__global__ __launch_bounds__(256) void k_layer1(const float* __restrict__ x, const float* __restrict__ dinv,
                                                const float* __restrict__ s1, const float* __restrict__ W1,
                                                const float* __restrict__ b1, float* __restrict__ h1, int N) {
  long long gid = (long long)blockIdx.x * blockDim.x + threadIdx.x;
  int n = (int)(gid >> 6), h = (int)(gid & 63);
  if (n < N) {
    float t = dinv[n] * (s1[n] + dinv[n] * x[n]);
    float v = t * W1[h] + b1[h];
    h1[(size_t)n * HID + h] = v > 0.0f ? v : 0.0f;
  }
}

// ---------------- WMMA GEMM: B[N,64] = A[N,64] @ W2[64,64] ----------------
// 256 threads = 8 waves; each wave computes a 16(M) x 64(N) tile, K=64 in
// steps of 4 via v_wmma_f32_16x16x4_f32 (4 col-tiles * 16 k-steps = 64 WMMAs).

__global__ __launch_bounds__(256) void k_gemm(const float* __restrict__ A, const float* __restrict__ W2g,
                                              float* __restrict__ B, int N) {
  __shared__ float w[HID * HID];
  for (int i = threadIdx.x; i < HID * HID; i += 256) w[i] = W2g[i];
  __syncthreads();

  int wave = threadIdx.x >> 5;
  int lane = threadIdx.x & 31;
  int l16  = lane & 15;
  int khalf = (lane >> 4) << 1;                 // 0 for lanes 0-15, 2 for lanes 16-31
  int rowBase = (blockIdx.x * 8 + wave) << 4;   // wave-uniform
  if (rowBase >= N) return;                     // uniform exit: EXEC stays all-ones below

  int row = rowBase + l16;
  if (row >= N) row = N - 1;                    // clamp load address only
  const float* arow = A + (size_t)row * HID;

  v8f c0 = {}, c1 = {}, c2 = {}, c3 = {};
  for (int k0 = 0; k0 < HID; k0 += 4) {
    // A frag 16x4 f32: lane L<16 holds A[M=L][k0],A[M=L][k0+1]; lane L+16 holds K=k0+2,k0+3
    v2f a;
    a[0] = arow[k0 + khalf];
    a[1] = arow[k0 + khalf + 1];
    // B frag 4x16 f32 per col-tile: lane L<16 holds B[k0][col],B[k0+1][col]; L+16: K+2,K+3
    const float* w0 = &w[(k0 + khalf) * HID];
    v2f b0, b1, b2, b3;
    b0[0] = w0[l16];        b0[1] = w0[HID + l16];
    b1[0] = w0[16 + l16];   b1[1] = w0[HID + 16 + l16];
    b2[0] = w0[32 + l16];   b2[1] = w0[HID + 32 + l16];
    b3[0] = w0[48 + l16];   b3[1] = w0[HID + 48 + l16];
    c0 = __builtin_amdgcn_wmma_f32_16x16x4_f32(false, a, false, b0, (short)0, c0, false, false);
    c1 = __builtin_amdgcn_wmma_f32_16x16x4_f32(false, a, false, b1, (short)0, c1, false, false);
    c2 = __builtin_amdgcn_wmma_f32_16x16x4_f32(false, a, false, b2, (short)0, c2, false, false);
    c3 = __builtin_amdgcn_wmma_f32_16x16x4_f32(false, a, false, b3, (short)0, c3, false, false);
  }

  // C/D layout: lane L<16 -> col=L, rows v (VGPR v); lane L>=16 -> col=L-16, rows v+8
  int rbase = rowBase + ((lane < 16) ? 0 : 8);
#pragma unroll
  for (int v = 0; v < 8; ++v) {
    int r = rbase + v;
    if (r < N) {
      float* o = B + (size_t)r * HID;
      o[l16]      = c0[v];
      o[16 + l16] = c1[v];
      o[32 + l16] = c2[v];
      o[48 + l16] = c3[v];
    }
  }
}

// ---------------- layer-2 scatter + finalize + pool ----------------

__global__ __launch_bounds__(256) void k_zero(float* __restrict__ p, long long n) {
  long long i = (long long)blockIdx.x * blockDim.x + threadIdx.x;
  if (i < n) p[i] = 0.0f;
}

// agg[dst][h] += dinv[src] * hw2[src][h]
__global__ __launch_bounds__(256) void k_scatter2(const int* __restrict__ src, const int* __restrict__ dst,
                                                  const float* __restrict__ dinv, const float* __restrict__ hw2,
                                                  float* __restrict__ agg, int E) {
  long long gid = (long long)blockIdx.x * blockDim.x + threadIdx.x;
  if (gid < (long long)E * HID) {
    int e = (int)(gid >> 6), h = (int)(gid & 63);
    int s = src[e];
    atomicAdd(&agg[(size_t)dst[e] * HID + h], dinv[s] * hw2[(size_t)s * HID + h]);
  }
}

// h2 = relu(dinv[n]*(agg + dinv[n]*hw2) + b2); accumulate into gsum/gcnt per graph
__global__ __launch_bounds__(256) void k_pool(const float* __restrict__ agg, const float* __restrict__ hw2,
                                              const float* __restrict__ dinv, const float* __restrict__ b2,
                                              const int* __restrict__ batch, float* __restrict__ gsum,
                                              float* __restrict__ gcnt, int N) {
  long long gid = (long long)blockIdx.x * blockDim.x + threadIdx.x;
  if (gid < (long long)N * HID) {
    int n = (int)(gid >> 6), h = (int)(gid & 63);
    float di = dinv[n];
    size_t idx = (size_t)n * HID + h;
    float v = di * (agg[idx] + di * hw2[idx]) + b2[h];
    v = v > 0.0f ? v : 0.0f;
    int g = batch[n];
    atomicAdd(&gsum[(size_t)g * HID + h], v);
    if (h == 0) atomicAdd(&gcnt[g], 1.0f);
  }
}

// out[g] = (sum_h gsum[g][h]*Wl[h]) / max(cnt,1) + bl
__global__ __launch_bounds__(256) void k_out(const float* __restrict__ gsum, const float* __restrict__ gcnt,
                                             const float* __restrict__ Wl, const float* __restrict__ bl,
                                             float* __restrict__ out, int G) {
  int g = blockIdx.x * blockDim.x + threadIdx.x;
  if (g < G) {
    float c = gcnt[g]; c = c > 1.0f ? c : 1.0f;
    float acc = 0.0f;
#pragma unroll
    for (int h = 0; h < HID; ++h) acc += gsum[(size_t)g * HID + h] * Wl[h];
    out[g] = acc / c + bl[0];
  }
}

// ---------------- host launcher ----------------

extern "C" void kernel_launch(void* const* d_in, const int* in_sizes, int n_in,
                              void* d_out, int out_size, void* d_ws, size_t ws_size,
                              hipStream_t stream) {
  const float* x   = (const float*)d_in[0];
  const float* W1  = (const float*)d_in[1];
  const float* b1  = (const float*)d_in[2];
  const float* W2  = (const float*)d_in[3];
  const float* b2  = (const float*)d_in[4];
  const float* Wl  = (const float*)d_in[5];
  const float* bl  = (const float*)d_in[6];
  const int*   ei  = (const int*)d_in[7];
  const int*   bat = (const int*)d_in[8];

  const int N = in_sizes[0];
  const int E = in_sizes[7] / 2;
  const int G = out_size;
  const int* src = ei;
  const int* dst = ei + E;

  float* ws = (float*)d_ws;
  size_t off = 0;
  float* bufA = ws + off; off += (size_t)N * HID;   // h1, then reused as agg2
  float* bufB = ws + off; off += (size_t)N * HID;   // hw2 = h1 @ W2
  float* deg  = ws + off; off += (size_t)N;
  float* dinv = ws + off; off += (size_t)N;
  float* s1   = ws + off; off += (size_t)N;
  float* gsum = ws + off; off += (size_t)G * HID;
  float* gcnt = ws + off; off += (size_t)G;
  (void)ws_size; (void)n_in;

  dim3 blk(256);
  auto gr = [](long long n) { return dim3((unsigned)((n + 255) / 256)); };

  long long initN = N > G * HID ? N : G * HID;
  k_init    <<<gr(initN),               blk, 0, stream>>>(deg, s1, gsum, gcnt, N, G);
  k_deg     <<<gr(E),                   blk, 0, stream>>>(dst, deg, E);
  k_dinv    <<<gr(N),                   blk, 0, stream>>>(deg, dinv, N);
  k_scatter1<<<gr(E),                   blk, 0, stream>>>(src, dst, x, dinv, s1, E);
  k_layer1  <<<gr((long long)N * HID),  blk, 0, stream>>>(x, dinv, s1, W1, b1, bufA, N);
  k_gemm    <<<dim3((N + 127) / 128),   blk, 0, stream>>>(bufA, W2, bufB, N);
  k_zero    <<<gr((long long)N * HID),  blk, 0, stream>>>(bufA, (long long)N * HID);
  k_scatter2<<<gr((long long)E * HID),  blk, 0, stream>>>(src, dst, dinv, bufB, bufA, E);
  k_pool    <<<gr((long long)N * HID),  blk, 0, stream>>>(bufA, bufB, dinv, b2, bat, gsum, gcnt, N);
  k_out     <<<gr(G),                   blk, 0, stream>>>(gsum, gcnt, Wl, bl, (float*)d_out, G);
}